// NIGConv_506806141219
// MI455X (gfx1250) — compile-verified
//
#include <hip/hip_runtime.h>
#include <hip/hip_bf16.h>
#include <math.h>
#include <stdint.h>

#define N_SRCN   50000
#define N_DSTN   50000
#define N_EDG    500000
#define IN_DIM   128
#define OUT_DIM  128
#define EDGE_DIM 11
#define BN_EPS   1e-5f

typedef float v2f __attribute__((ext_vector_type(2)));
typedef float v8f __attribute__((ext_vector_type(8)));

// Monotone float <-> signed-int encoding (involution), so signed atomicMax
// on the encoded value == float max, including negatives.
__device__ __forceinline__ int f32_ord(float f) {
  int i = __float_as_int(f);
  return i >= 0 ? i : (i ^ 0x7fffffff);
}
__device__ __forceinline__ float ord_f32(int i) {
  return __int_as_float(i >= 0 ? i : (i ^ 0x7fffffff));
}

// ---------------------------------------------------------------------------
// 0) Fold W_dst@aw_d -> wv_d[128], W_edge@aw_e -> wv_e[11], bias dots -> bb[2]
// ---------------------------------------------------------------------------
__global__ __launch_bounds__(128) void prep_kernel(
    const float* __restrict__ W_dst, const float* __restrict__ b_dst,
    const float* __restrict__ W_edge, const float* __restrict__ b_edge,
    const float* __restrict__ attn_w,
    float* __restrict__ wv_d, float* __restrict__ wv_e, float* __restrict__ bb)
{
  const int t = threadIdx.x;
  const float* aw_d = attn_w + OUT_DIM;
  const float* aw_e = attn_w + 2 * OUT_DIM;
  float s = 0.f;
  for (int j = 0; j < OUT_DIM; ++j) s += W_dst[t * OUT_DIM + j] * aw_d[j];
  wv_d[t] = s;
  if (t < EDGE_DIM) {
    float s2 = 0.f;
    for (int j = 0; j < OUT_DIM; ++j) s2 += W_edge[t * OUT_DIM + j] * aw_e[j];
    wv_e[t] = s2;
  }
  if (t == 0) {
    float bd = 0.f, be = 0.f;
    for (int j = 0; j < OUT_DIM; ++j) { bd += b_dst[j] * aw_d[j]; be += b_edge[j] * aw_e[j]; }
    bb[0] = bd; bb[1] = be;
  }
}

// ---------------------------------------------------------------------------
// 1) Zero accumulators (h_new lives in d_out), seed segment max with -inf
// ---------------------------------------------------------------------------
__global__ __launch_bounds__(256) void init_kernel(
    float* __restrict__ h_new, int* __restrict__ segmax, float* __restrict__ denom,
    float* __restrict__ colsum, float* __restrict__ colsumsq)
{
  const int idx = blockIdx.x * blockDim.x + threadIdx.x;
  const int total = N_DSTN * OUT_DIM;
  if (idx < total)  h_new[idx] = 0.f;
  if (idx < N_DSTN) { segmax[idx] = f32_ord(-INFINITY); denom[idx] = 0.f; }
  if (idx < OUT_DIM) { colsum[idx] = 0.f; colsumsq[idx] = 0.f; }
}

// ---------------------------------------------------------------------------
// 2) hs = src_feat @ W_src + b_src  via V_WMMA_F32_16X16X4_F32
//    fused: s_dot[row] = hs[row,:] . aw_s
//    Block = 128 threads (4 waves); each wave owns a 16-row tile and all
//    8 N-tiles (8 x v8f accumulators).
//    W_src staged into LDS with CDNA5 async DMA:
//      global_load_async_to_lds_b128 (GVS form), ASYNCcnt-tracked,
//      then s_wait_asynccnt 0 + workgroup barrier.
// ---------------------------------------------------------------------------
__global__ __launch_bounds__(128) void hs_gemm_kernel(
    const float* __restrict__ src_feat, const float* __restrict__ W,
    const float* __restrict__ bias, const float* __restrict__ aw,
    float* __restrict__ hs, float* __restrict__ s_dot, int n_rows)
{
  __shared__ float Wlds[IN_DIM * OUT_DIM];  // 64 KB of the 320 KB WGP LDS
  const int tid = threadIdx.x;

  { // async DMA staging of W (row-major 128x128 = 64KB), 16B per issue.
    // LDS dest operand = LDS-relative byte offset (low 32 bits of the
    // generic pointer to a __shared__ object).
    const uint32_t lds_base = (uint32_t)(uintptr_t)(&Wlds[0]);
    uint32_t dst  = lds_base + (uint32_t)tid * 16u;
    uint32_t voff = (uint32_t)tid * 16u;
    for (int i = 0; i < 32; ++i) {  // 128 thr * 32 iter * 16B = 64KB
      asm volatile("global_load_async_to_lds_b128 %0, %1, %2"
                   :
                   : "v"(dst), "v"(voff), "s"(W)
                   : "memory");
      dst  += 128u * 16u;
      voff += 128u * 16u;
    }
    asm volatile("s_wait_asynccnt 0" ::: "memory");
  }
  __syncthreads();

  const int wave  = tid >> 5;
  const int lane  = tid & 31;
  const int m     = lane & 15;   // A-frag: row within tile
  const int khalf = lane >> 4;   // A/B frag: which K-pair
  const int tile_row = (blockIdx.x * 4 + wave) * 16;

  v8f zero = {};
  v8f acc[8];
#pragma unroll
  for (int i = 0; i < 8; ++i) acc[i] = zero;

  int arow = tile_row + m;
  if (arow >= n_rows) arow = n_rows - 1;          // clamp; guarded on store
  const float* __restrict__ Arow = src_feat + (size_t)arow * IN_DIM;

  for (int k0 = 0; k0 < IN_DIM; k0 += 4) {
    v2f a;
    a.x = Arow[k0 + 2 * khalf + 0];
    a.y = Arow[k0 + 2 * khalf + 1];
    const int n16 = lane & 15;
#pragma unroll
    for (int nt = 0; nt < 8; ++nt) {
      v2f b;
      const int n = nt * 16 + n16;
      b.x = Wlds[(k0 + 2 * khalf + 0) * OUT_DIM + n];
      b.y = Wlds[(k0 + 2 * khalf + 1) * OUT_DIM + n];
      acc[nt] = __builtin_amdgcn_wmma_f32_16x16x4_f32(
          /*neg_a=*/false, a, /*neg_b=*/false, b,
          /*c_mod=*/(short)0, acc[nt], /*reuse_a=*/false, /*reuse_b=*/false);
    }
  }

  // C/D layout: lane -> (n = lane&15, mhalf = lane>>4), vgpr r -> row r+8*mhalf
  const int n16   = lane & 15;
  const int mhalf = lane >> 4;
  float bias_r[8], aw_r[8];
#pragma unroll
  for (int nt = 0; nt < 8; ++nt) {
    bias_r[nt] = bias[nt * 16 + n16];
    aw_r[nt]   = aw[nt * 16 + n16];
  }
#pragma unroll
  for (int r = 0; r < 8; ++r) {
    const int orow = tile_row + r + 8 * mhalf;
    float vals[8];
    float pdot = 0.f;
#pragma unroll
    for (int nt = 0; nt < 8; ++nt) {
      const float v = acc[nt][r] + bias_r[nt];
      vals[nt] = v;
      pdot += v * aw_r[nt];
    }
    // reduce across the 16 lanes sharing this output row (wave32 shuffles)
    pdot += __shfl_xor(pdot, 1, 32);
    pdot += __shfl_xor(pdot, 2, 32);
    pdot += __shfl_xor(pdot, 4, 32);
    pdot += __shfl_xor(pdot, 8, 32);
    if (orow < n_rows) {
#pragma unroll
      for (int nt = 0; nt < 8; ++nt)
        hs[(size_t)orow * OUT_DIM + nt * 16 + n16] = vals[nt];
      if (n16 == 0) s_dot[orow] = pdot;
    }
  }
}

// ---------------------------------------------------------------------------
// 3) d_dot = dst_feat @ wv_d + bb[0]   (one wave per row)
// ---------------------------------------------------------------------------
__global__ __launch_bounds__(256) void dvec_kernel(
    const float* __restrict__ dst_feat, const float* __restrict__ wv_d,
    const float* __restrict__ bb, float* __restrict__ d_dot, int n_rows)
{
  const int gw   = (blockIdx.x * blockDim.x + threadIdx.x) >> 5;
  const int lane = threadIdx.x & 31;
  if (gw >= n_rows) return;
  const float* __restrict__ row = dst_feat + (size_t)gw * IN_DIM;
  float s = 0.f;
#pragma unroll
  for (int j = 0; j < 4; ++j) s += row[lane + 32 * j] * wv_d[lane + 32 * j];
  s += __shfl_xor(s, 16, 32);
  s += __shfl_xor(s, 8, 32);
  s += __shfl_xor(s, 4, 32);
  s += __shfl_xor(s, 2, 32);
  s += __shfl_xor(s, 1, 32);
  if (lane == 0) d_dot[gw] = s + bb[0];
}

// ---------------------------------------------------------------------------
// 4) per-edge logit + segment max (ordered-int atomicMax)
// ---------------------------------------------------------------------------
__global__ __launch_bounds__(256) void edgeA_kernel(
    const float* __restrict__ edge_feats, const int* __restrict__ src_idx,
    const int* __restrict__ dst_idx, const float* __restrict__ s_dot,
    const float* __restrict__ d_dot, const float* __restrict__ wv_e,
    const float* __restrict__ bb, const float* __restrict__ attn_b,
    float* __restrict__ a_buf, int* __restrict__ segmax)
{
  const int e = blockIdx.x * blockDim.x + threadIdx.x;
  if (e >= N_EDG) return;
  const float* __restrict__ r = edge_feats + (size_t)e * EDGE_DIM;
  float ed = bb[1];
#pragma unroll
  for (int j = 0; j < EDGE_DIM; ++j) ed += r[j] * wv_e[j];
  const int d = dst_idx[e];
  const float a = s_dot[src_idx[e]] + d_dot[d] + ed + attn_b[0];
  a_buf[e] = a;
  atomicMax(&segmax[d], f32_ord(a));
}

// ---------------------------------------------------------------------------
// 5) a_exp + segment sum
// ---------------------------------------------------------------------------
__global__ __launch_bounds__(256) void edgeB_kernel(
    const int* __restrict__ dst_idx, const int* __restrict__ segmax,
    float* __restrict__ a_buf, float* __restrict__ denom)
{
  const int e = blockIdx.x * blockDim.x + threadIdx.x;
  if (e >= N_EDG) return;
  const int d = dst_idx[e];
  const float x = __expf(a_buf[e] - ord_f32(segmax[d]));
  a_buf[e] = x;
  atomicAdd(&denom[d], x);
}

// ---------------------------------------------------------------------------
// 6) message scatter: h_new[dst] += attn * hs[src]   (one wave per edge)
// ---------------------------------------------------------------------------
__global__ __launch_bounds__(256) void msg_kernel(
    const int* __restrict__ src_idx, const int* __restrict__ dst_idx,
    const float* __restrict__ a_buf, const float* __restrict__ denom,
    const float* __restrict__ hs, float* __restrict__ h_new)
{
  const int e    = (blockIdx.x * blockDim.x + threadIdx.x) >> 5;
  const int lane = threadIdx.x & 31;
  if (e >= N_EDG) return;
  const int s = src_idx[e];
  const int d = dst_idx[e];
  const float attn = a_buf[e] / denom[d];
  const float* __restrict__ hrow = hs + (size_t)s * OUT_DIM;
  float* __restrict__ orow = h_new + (size_t)d * OUT_DIM;
  __builtin_prefetch(hrow, 0, 0);  // global_prefetch_b8 for the irregular gather
#pragma unroll
  for (int j = 0; j < 4; ++j) {
    const int c = lane + 32 * j;
    atomicAdd(&orow[c], attn * hrow[c]);
  }
}

// ---------------------------------------------------------------------------
// 7) BN column statistics: 256 rows per block, 1 atomicAdd pair per column
// ---------------------------------------------------------------------------
__global__ __launch_bounds__(128) void bnstats_kernel(
    const float* __restrict__ h_new, float* __restrict__ colsum,
    float* __restrict__ colsumsq)
{
  const int c  = threadIdx.x;                 // column 0..127
  const int r0 = blockIdx.x * 256;
  float s = 0.f, s2 = 0.f;
  const int rend = min(r0 + 256, N_DSTN);
  for (int r = r0; r < rend; ++r) {
    const float v = h_new[(size_t)r * OUT_DIM + c];
    s += v; s2 += v * v;
  }
  atomicAdd(&colsum[c], s);
  atomicAdd(&colsumsq[c], s2);
}

// ---------------------------------------------------------------------------
// 8) BN + leaky ReLU (in place on d_out)
// ---------------------------------------------------------------------------
__global__ __launch_bounds__(256) void finalize_kernel(
    float* __restrict__ h, const float* __restrict__ colsum,
    const float* __restrict__ colsumsq, const float* __restrict__ gamma,
    const float* __restrict__ beta, const float* __restrict__ alpha)
{
  const int idx = blockIdx.x * blockDim.x + threadIdx.x;
  if (idx >= N_DSTN * OUT_DIM) return;
  const int c = idx & (OUT_DIM - 1);
  const float inv_n = 1.0f / (float)N_DSTN;
  const float mean  = colsum[c] * inv_n;
  const float var   = colsumsq[c] * inv_n - mean * mean;
  const float hb = (h[idx] - mean) * rsqrtf(var + BN_EPS) * gamma[c] + beta[c];
  h[idx] = hb >= 0.f ? hb : alpha[0] * hb;
}

// ---------------------------------------------------------------------------
extern "C" void kernel_launch(void* const* d_in, const int* in_sizes, int n_in,
                              void* d_out, int out_size, void* d_ws, size_t ws_size,
                              hipStream_t stream) {
  (void)in_sizes; (void)n_in; (void)out_size; (void)ws_size;
  const float* src_feat   = (const float*)d_in[0];
  const float* dst_feat   = (const float*)d_in[1];
  const float* edge_feats = (const float*)d_in[2];
  const int*   src_idx    = (const int*)d_in[3];
  const int*   dst_idx    = (const int*)d_in[4];
  const float* W_src      = (const float*)d_in[5];
  const float* b_src      = (const float*)d_in[6];
  const float* W_dst      = (const float*)d_in[7];
  const float* b_dst      = (const float*)d_in[8];
  const float* W_edge     = (const float*)d_in[9];
  const float* b_edge     = (const float*)d_in[10];
  const float* attn_w     = (const float*)d_in[11];
  const float* attn_b     = (const float*)d_in[12];
  const float* gamma      = (const float*)d_in[13];
  const float* beta       = (const float*)d_in[14];
  const float* alpha      = (const float*)d_in[15];

  float* h_new = (float*)d_out;  // accumulate h_new in d_out, finalize in place

  // workspace carve (256B aligned)
  size_t off = 0;
  auto carve = [&](size_t bytes) -> char* {
    off = (off + 255) & ~(size_t)255;
    char* p = (char*)d_ws + off;
    off += bytes;
    return p;
  };
  float* hs      = (float*)carve((size_t)N_SRCN * OUT_DIM * sizeof(float)); // 25.6MB
  float* s_dot   = (float*)carve((size_t)N_SRCN * sizeof(float));
  float* d_dot   = (float*)carve((size_t)N_DSTN * sizeof(float));
  float* a_buf   = (float*)carve((size_t)N_EDG * sizeof(float));
  int*   segmax  = (int*)  carve((size_t)N_DSTN * sizeof(int));
  float* denom   = (float*)carve((size_t)N_DSTN * sizeof(float));
  float* wv_d    = (float*)carve(OUT_DIM * sizeof(float));
  float* wv_e    = (float*)carve(EDGE_DIM * sizeof(float));
  float* bb      = (float*)carve(2 * sizeof(float));
  float* colsum  = (float*)carve(OUT_DIM * sizeof(float));
  float* colsumsq= (float*)carve(OUT_DIM * sizeof(float));

  // 0) fold weights
  prep_kernel<<<1, 128, 0, stream>>>(W_dst, b_dst, W_edge, b_edge, attn_w,
                                     wv_d, wv_e, bb);
  // 1) init accumulators
  {
    const int total = N_DSTN * OUT_DIM;
    init_kernel<<<(total + 255) / 256, 256, 0, stream>>>(h_new, segmax, denom,
                                                         colsum, colsumsq);
  }
  // 2) hs GEMM (WMMA f32 16x16x4), 64 rows/block, async-DMA W staging
  hs_gemm_kernel<<<(N_SRCN + 63) / 64, 128, 0, stream>>>(
      src_feat, W_src, b_src, attn_w /* aw_s = attn_w[0:128] */, hs, s_dot, N_SRCN);
  // 3) d_dot mat-vec, 8 waves/block
  dvec_kernel<<<(N_DSTN * 32 + 255) / 256, 256, 0, stream>>>(dst_feat, wv_d, bb,
                                                             d_dot, N_DSTN);
  // 4) edge logits + segment max
  edgeA_kernel<<<(N_EDG + 255) / 256, 256, 0, stream>>>(
      edge_feats, src_idx, dst_idx, s_dot, d_dot, wv_e, bb, attn_b, a_buf, segmax);
  // 5) exp + segment sum
  edgeB_kernel<<<(N_EDG + 255) / 256, 256, 0, stream>>>(dst_idx, segmax, a_buf, denom);
  // 6) message scatter, one wave per edge
  msg_kernel<<<(N_EDG * 32 + 255) / 256, 256, 0, stream>>>(src_idx, dst_idx, a_buf,
                                                           denom, hs, h_new);
  // 7) BN stats
  bnstats_kernel<<<(N_DSTN + 255) / 256, 128, 0, stream>>>(h_new, colsum, colsumsq);
  // 8) BN + leaky ReLU in place
  finalize_kernel<<<(N_DSTN * OUT_DIM + 255) / 256, 256, 0, stream>>>(
      h_new, colsum, colsumsq, gamma, beta, alpha);
}